// HGNNMLP_57492432224277
// MI455X (gfx1250) — compile-verified
//
#include <hip/hip_runtime.h>
#include <hip/hip_bf16.h>

// ---------------------------------------------------------------------------
// HGNN forward, algebraically folded (only out_email is live in the reference).
//   out = relu( x_email@Wfe + scatter_ue(x_url@Wfu + bu) + scatter_se(x_s*Wfs + bs) + be ) @ Wc + bc
// Wfe = W_email@(Wroot_ue+Wroot_se)  -- stored pre-swizzled in the LDS-ready
// K-pair-interleaved layout so the GEMM stages it with identity
// GLOBAL_LOAD_ASYNC_TO_LDS_B128 copies and reads B-frags as contiguous b64.
// Big GEMM: V_WMMA_F32_16X16X4_F32 (fp32 kept: kernel is HBM-bound).
// ---------------------------------------------------------------------------

#define N_EMAIL  100000
#define N_URL    400000
#define N_SENDER 50000
#define HID      128
#define KDIM     768
#define E_UE     1500000
#define E_SE     800000

#define KC      64                  // K-chunk per LDS slab
#define NCHUNK  (KDIM / KC)         // 12
#define PAIRROW 288                 // floats per K-pair row: col*2+(k&1) in [0,256), pad to 288
                                    // 288%64 == 32 -> lanes 0-15/16-31 on disjoint bank halves
#define CHUNKF  ((KC / 2) * PAIRROW)  // 9216 floats = 36,864 B per slab

typedef float v2f __attribute__((ext_vector_type(2)));
typedef float v8f __attribute__((ext_vector_type(8)));

// workspace layout (float offsets)
#define WS_WFE    0                  // NCHUNK*CHUNKF = 110592 (swizzled Wfe)
#define WS_BIASE  110592             // 128
#define WS_WFU    110720             // 8*128 (bias_u immediately after -> contiguous 1152)
#define WS_BIASU  111744             // 128
#define WS_WFS    111872             // 128   (bias_s immediately after -> contiguous 256)
#define WS_BIASS  112000             // 128
#define WS_ACC    112128             // N_EMAIL*128

// ---- weight fusion (writes Wfe directly in swizzled slab layout) ---------
__global__ __launch_bounds__(256) void k_fuse_email(
    const float* __restrict__ W_email, const float* __restrict__ Wroot_ue,
    const float* __restrict__ Wroot_se, float* __restrict__ Wfe) {
  int idx = blockIdx.x * 256 + threadIdx.x;      // 768*128 threads
  int r = idx >> 7, c = idx & 127;               // r = K row, c = N col
  float s = 0.f;
  for (int k = 0; k < HID; ++k)
    s += W_email[r * HID + k] * (Wroot_ue[k * HID + c] + Wroot_se[k * HID + c]);
  int ch = r >> 6, kl = r & 63;                  // chunk, K-within-chunk
  Wfe[ch * CHUNKF + (kl >> 1) * PAIRROW + c * 2 + (kl & 1)] = s;
}

__global__ __launch_bounds__(128) void k_fuse_small(
    const float* __restrict__ b_email, const float* __restrict__ brel_ue,
    const float* __restrict__ brel_se, const float* __restrict__ Wroot_ue,
    const float* __restrict__ Wroot_se, const float* __restrict__ W_url,
    const float* __restrict__ b_url, const float* __restrict__ Wrel_ue,
    const float* __restrict__ W_sender, const float* __restrict__ b_sender,
    const float* __restrict__ Wrel_se, float* __restrict__ ws) {
  int c = threadIdx.x;  // 128 threads
  float be = brel_ue[c] + brel_se[c];
  for (int k = 0; k < HID; ++k)
    be += b_email[k] * (Wroot_ue[k * HID + c] + Wroot_se[k * HID + c]);
  ws[WS_BIASE + c] = be;
  for (int r = 0; r < 8; ++r) {
    float s = 0.f;
    for (int m = 0; m < HID; ++m) s += W_url[r * HID + m] * Wrel_ue[m * HID + c];
    ws[WS_WFU + r * HID + c] = s;
  }
  float bu = 0.f, wfs = 0.f, bs = 0.f;
  for (int m = 0; m < HID; ++m) {
    bu  += b_url[m]    * Wrel_ue[m * HID + c];
    wfs += W_sender[m] * Wrel_se[m * HID + c];
    bs  += b_sender[m] * Wrel_se[m * HID + c];
  }
  ws[WS_BIASU + c] = bu;
  ws[WS_WFS + c]   = wfs;
  ws[WS_BIASS + c] = bs;
}

// ---- accumulator init (fused bias broadcast) -----------------------------
__global__ __launch_bounds__(256) void k_init_acc(const float* __restrict__ biasE,
                                                  float* __restrict__ acc) {
  int i = blockIdx.x * 256 + threadIdx.x;  // exactly N_EMAIL*HID
  acc[i] = biasE[i & 127];
}

// ---- edge scatter: url -> email (message computed on the fly, K=8) -------
__global__ __launch_bounds__(256) void k_scatter_url(
    const float* __restrict__ x_url, const int* __restrict__ src,
    const int* __restrict__ dst, const float* __restrict__ ws,
    float* __restrict__ acc) {
  __shared__ float sW[9 * HID];  // Wfu[8][128] ++ bias_u[128] (contiguous in ws)
  for (int i = threadIdx.x; i < 9 * HID; i += 256) sW[i] = ws[WS_WFU + i];
  __syncthreads();
  int edge = blockIdx.x * 8 + (threadIdx.x >> 5);  // one wave32 per edge
  if (edge >= E_UE) return;
  int lane = threadIdx.x & 31;
  int s = src[edge], d = dst[edge];
  float4 xa = *(const float4*)(x_url + s * 8);
  float4 xb = *(const float4*)(x_url + s * 8 + 4);
  float xv[8] = {xa.x, xa.y, xa.z, xa.w, xb.x, xb.y, xb.z, xb.w};
  int c0 = lane * 4;
  float4 m = *(const float4*)&sW[8 * HID + c0];  // bias_u
#pragma unroll
  for (int k = 0; k < 8; ++k) {
    float4 w = *(const float4*)&sW[k * HID + c0];
    m.x += xv[k] * w.x; m.y += xv[k] * w.y;
    m.z += xv[k] * w.z; m.w += xv[k] * w.w;
  }
  float* ap = acc + d * HID + c0;  // L2-resident accumulator (51 MB < 192 MB L2)
  atomicAdd(ap + 0, m.x); atomicAdd(ap + 1, m.y);
  atomicAdd(ap + 2, m.z); atomicAdd(ap + 3, m.w);
}

// ---- edge scatter: sender -> email (rank-1 message) ----------------------
__global__ __launch_bounds__(256) void k_scatter_sender(
    const float* __restrict__ x_sender, const int* __restrict__ src,
    const int* __restrict__ dst, const float* __restrict__ ws,
    float* __restrict__ acc) {
  __shared__ float sW[2 * HID];  // Wfs[128] ++ bias_s[128]
  for (int i = threadIdx.x; i < 2 * HID; i += 256) sW[i] = ws[WS_WFS + i];
  __syncthreads();
  int edge = blockIdx.x * 8 + (threadIdx.x >> 5);
  if (edge >= E_SE) return;
  int lane = threadIdx.x & 31;
  float xv = x_sender[src[edge]];
  int d = dst[edge];
  int c0 = lane * 4;
  float4 w = *(const float4*)&sW[c0];
  float4 b = *(const float4*)&sW[HID + c0];
  float* ap = acc + d * HID + c0;
  atomicAdd(ap + 0, b.x + xv * w.x); atomicAdd(ap + 1, b.y + xv * w.y);
  atomicAdd(ap + 2, b.z + xv * w.z); atomicAdd(ap + 3, b.w + xv * w.w);
}

// ---- big GEMM: acc = relu(acc + x_email @ Wfe), WMMA f32 16x16x4 ---------
__global__ __launch_bounds__(256) void k_gemm_relu(
    const float* __restrict__ X, const float* __restrict__ WfeI,
    float* __restrict__ acc) {
  __shared__ float sB[CHUNKF];  // 36,864 B, identical layout to one global slab
  const int tid  = threadIdx.x;
  const int wave = tid >> 5;
  const int lane = tid & 31;
  const int ln   = lane & 15;
  const int kh   = (lane >> 4) << 1;        // f32 A-frag: lanes 16-31 carry K+2,K+3
  const int mBase = blockIdx.x * 128 + wave * 16;
  const int row   = mBase + ln;
  const int rowc  = row < N_EMAIL ? row : N_EMAIL - 1;  // clamp keeps EXEC all-1s for WMMA
  const float* xrow = X + (long long)rowc * KDIM;

  v8f zero = {};
  v8f cacc[8];
#pragma unroll
  for (int i = 0; i < 8; ++i) cacc[i] = zero;

  for (int ch = 0; ch < NCHUNK; ++ch) {
    const float* slab = WfeI + ch * CHUNKF;
    __syncthreads();  // all DS reads of previous slab retired before new writes
    // identity copy: global slab -> LDS via async tensor-path copies
    // (GLOBAL_LOAD_ASYNC_TO_LDS_B128: no VGPR data, tracked by ASYNCcnt)
    for (int it = tid; it < CHUNKF / 4; it += 256) {  // 9 iters/thread, uniform
      const float* gsrc = slab + it * 4;
      unsigned ldst = (unsigned)(uintptr_t)(&sB[it * 4]);
      asm volatile("global_load_async_to_lds_b128 %0, %1, off"
                   :: "v"(ldst), "v"(gsrc)
                   : "memory");
    }
    if (ch + 1 < NCHUNK)  // hint next slab toward L2 (global_prefetch_b8)
      __builtin_prefetch(slab + CHUNKF + tid * 36, 0, 1);
    asm volatile("s_wait_asynccnt 0x0" ::: "memory");
    __syncthreads();

    for (int k0 = 0; k0 < KC; k0 += 4) {
      float2 av = *(const float2*)(xrow + ch * KC + k0 + kh);
      v2f a; a[0] = av.x; a[1] = av.y;
      // preload all 8 B-frags (contiguous K-pairs -> ds_load_2addr_b64, no
      // repacking), then fire 8 independent WMMA accumulation chains
      const float* bp = &sB[((k0 + kh) >> 1) * PAIRROW + ln * 2];
      v2f bf[8];
#pragma unroll
      for (int nt = 0; nt < 8; ++nt)
        bf[nt] = *(const v2f*)(bp + nt * 32);
#pragma unroll
      for (int nt = 0; nt < 8; ++nt)
        cacc[nt] = __builtin_amdgcn_wmma_f32_16x16x4_f32(
            false, a, false, bf[nt], (short)0, cacc[nt], false, false);
    }
  }

  // epilogue: add scattered messages + bias (already in acc), relu, in-place store
  const int mOff = (lane >> 4) << 3;  // C layout: lanes 16-31 -> M+8
#pragma unroll
  for (int nt = 0; nt < 8; ++nt) {
    int col = nt * 16 + ln;
#pragma unroll
    for (int v = 0; v < 8; ++v) {
      int r = mBase + v + mOff;
      if (r < N_EMAIL) {
        float g = acc[r * HID + col] + cacc[nt][v];
        acc[r * HID + col] = g > 0.f ? g : 0.f;
      }
    }
  }
}

// ---- classifier: out[n] = relu_row @ Wc + bc (N=2), wave-per-row ---------
__global__ __launch_bounds__(256) void k_classifier(
    const float* __restrict__ acc, const float* __restrict__ Wc,
    const float* __restrict__ bc, float* __restrict__ out) {
  int n = blockIdx.x * 8 + (threadIdx.x >> 5);
  if (n >= N_EMAIL) return;
  int lane = threadIdx.x & 31;
  float4 a  = *(const float4*)(acc + n * HID + lane * 4);
  float4 wA = *(const float4*)(Wc + lane * 8);      // {w(j,0),w(j,1),w(j+1,0),w(j+1,1)}
  float4 wB = *(const float4*)(Wc + lane * 8 + 4);
  float p0 = a.x * wA.x + a.y * wA.z + a.z * wB.x + a.w * wB.z;
  float p1 = a.x * wA.y + a.y * wA.w + a.z * wB.y + a.w * wB.w;
  for (int off = 16; off > 0; off >>= 1) {
    p0 += __shfl_down(p0, off, 32);
    p1 += __shfl_down(p1, off, 32);
  }
  if (lane == 0) {
    out[n * 2 + 0] = p0 + bc[0];
    out[n * 2 + 1] = p1 + bc[1];
  }
}

// ---------------------------------------------------------------------------
extern "C" void kernel_launch(void* const* d_in, const int* in_sizes, int n_in,
                              void* d_out, int out_size, void* d_ws, size_t ws_size,
                              hipStream_t stream) {
  (void)in_sizes; (void)n_in; (void)out_size; (void)ws_size;
  const float* x_email  = (const float*)d_in[0];
  const float* x_url    = (const float*)d_in[1];
  const float* x_sender = (const float*)d_in[2];
  const float* W_email  = (const float*)d_in[3];
  const float* b_email  = (const float*)d_in[4];
  const float* W_url    = (const float*)d_in[5];
  const float* b_url    = (const float*)d_in[6];
  const float* W_sender = (const float*)d_in[7];
  const float* b_sender = (const float*)d_in[8];
  const float* Wrel_ue  = (const float*)d_in[9];
  const float* brel_ue  = (const float*)d_in[10];
  const float* Wroot_ue = (const float*)d_in[11];
  const float* Wrel_se  = (const float*)d_in[12];
  const float* brel_se  = (const float*)d_in[13];
  const float* Wroot_se = (const float*)d_in[14];
  // inputs 15..20 (eu/es edge-type weights) are dead in the reference output
  const float* Wc = (const float*)d_in[21];
  const float* bc = (const float*)d_in[22];
  const int* src_ue = (const int*)d_in[23];
  const int* dst_ue = (const int*)d_in[24];
  const int* src_se = (const int*)d_in[25];
  const int* dst_se = (const int*)d_in[26];

  float* ws  = (float*)d_ws;
  float* out = (float*)d_out;
  float* Wfe = ws + WS_WFE;
  float* acc = ws + WS_ACC;

  hipLaunchKernelGGL(k_fuse_email, dim3((KDIM * HID) / 256), dim3(256), 0, stream,
                     W_email, Wroot_ue, Wroot_se, Wfe);
  hipLaunchKernelGGL(k_fuse_small, dim3(1), dim3(128), 0, stream,
                     b_email, brel_ue, brel_se, Wroot_ue, Wroot_se,
                     W_url, b_url, Wrel_ue, W_sender, b_sender, Wrel_se, ws);
  hipLaunchKernelGGL(k_init_acc, dim3((N_EMAIL * HID) / 256), dim3(256), 0, stream,
                     ws + WS_BIASE, acc);
  hipLaunchKernelGGL(k_scatter_url, dim3((E_UE + 7) / 8), dim3(256), 0, stream,
                     x_url, src_ue, dst_ue, ws, acc);
  hipLaunchKernelGGL(k_scatter_sender, dim3((E_SE + 7) / 8), dim3(256), 0, stream,
                     x_sender, src_se, dst_se, ws, acc);
  hipLaunchKernelGGL(k_gemm_relu, dim3((N_EMAIL + 127) / 128), dim3(256), 0, stream,
                     x_email, Wfe, acc);
  hipLaunchKernelGGL(k_classifier, dim3((N_EMAIL + 7) / 8), dim3(256), 0, stream,
                     acc, Wc, bc, out);
}